// Instance_loss_15839839388116
// MI455X (gfx1250) — compile-verified
//
#include <hip/hip_runtime.h>
#include <hip/hip_bf16.h>
#include <math.h>

// Problem constants (fixed by the reference)
#define B 8
#define C 4
#define H 512
#define W 1024
#define NPIX (H * W)          // 524288 pixels per image
#define NPIX4 (NPIX / 4)      // 131072 float4 groups per channel plane
#define M 9                   // labels 0..8 (0 = background)
#define DELTA_VAR 0.5f
#define EPS_L 1e-8f
#define BPB 64                // blocks per batch image for the streaming passes
#define TPB 256               // threads per block

typedef float v2f __attribute__((ext_vector_type(2)));
typedef float v8f __attribute__((ext_vector_type(8)));
typedef unsigned int u32x4 __attribute__((ext_vector_type(4)));
typedef int i32x4 __attribute__((ext_vector_type(4)));
typedef int i32x8 __attribute__((ext_vector_type(8)));

// ---------------- workspace layout (floats) ----------------
// sums   : [0,   288)   b*M*4
// counts : [288, 360)   b*M
// hsum   : [360, 432)   b*M
// means  : [432, 720)   b*M*4   (16B aligned: 432*4 = 1728)
#define WS_SUMS   0
#define WS_COUNTS 288
#define WS_HSUM   360
#define WS_MEANS  432
#define WS_ZERO_N 432

__global__ void il_zero_ws(float* ws) {
    for (int i = threadIdx.x; i < WS_ZERO_N; i += blockDim.x) ws[i] = 0.0f;
}

// Per-pixel predicated accumulation into 45 register bins.
#define ACC_PIX(lv, v0, v1, v2, v3)                                     \
    _Pragma("unroll")                                                   \
    for (int q = 0; q < M; ++q) {                                       \
        const float msk = ((lv) == q) ? 1.0f : 0.0f;                    \
        cnt[q]   += msk;                                                \
        sm[q][0] += msk * (v0);                                         \
        sm[q][1] += msk * (v1);                                         \
        sm[q][2] += msk * (v2);                                         \
        sm[q][3] += msk * (v3);                                         \
    }

// Pass 1: per-(batch,label) channel sums + counts. b128 loads (4 px/thread/iter),
// register accumulators, LDS block reduce, one global atomic per bin per block.
__global__ void il_pass1(const float* __restrict__ x, const int* __restrict__ lab,
                         float* __restrict__ sums, float* __restrict__ counts) {
    const int bi  = blockIdx.x / BPB;
    const int blk = blockIdx.x % BPB;

    __shared__ float s_acc[M * 5];
    for (int i = threadIdx.x; i < M * 5; i += blockDim.x) s_acc[i] = 0.0f;
    __syncthreads();

    float cnt[M];
    float sm[M][C];
#pragma unroll
    for (int q = 0; q < M; ++q) {
        cnt[q] = 0.0f;
#pragma unroll
        for (int ch = 0; ch < C; ++ch) sm[q][ch] = 0.0f;
    }

    const float4* xb4 = (const float4*)(x + (size_t)bi * C * NPIX);
    const int4*   lb4 = (const int4*)(lab + (size_t)bi * NPIX);
    const int stride = BPB * TPB;

    for (int p = blk * TPB + threadIdx.x; p < NPIX4; p += stride) {
        __builtin_prefetch(lb4 + p + stride, 0, 0);   // -> global_prefetch_b8
        __builtin_prefetch(xb4 + p + stride, 0, 0);
        const int4   l4 = lb4[p];
        const float4 c0 = xb4[p];
        const float4 c1 = xb4[p + NPIX4];
        const float4 c2 = xb4[p + 2 * NPIX4];
        const float4 c3 = xb4[p + 3 * NPIX4];
        ACC_PIX(l4.x, c0.x, c1.x, c2.x, c3.x)
        ACC_PIX(l4.y, c0.y, c1.y, c2.y, c3.y)
        ACC_PIX(l4.z, c0.z, c1.z, c2.z, c3.z)
        ACC_PIX(l4.w, c0.w, c1.w, c2.w, c3.w)
    }

#pragma unroll
    for (int q = 0; q < M; ++q) {
        atomicAdd(&s_acc[q * 5 + 0], sm[q][0]);
        atomicAdd(&s_acc[q * 5 + 1], sm[q][1]);
        atomicAdd(&s_acc[q * 5 + 2], sm[q][2]);
        atomicAdd(&s_acc[q * 5 + 3], sm[q][3]);
        atomicAdd(&s_acc[q * 5 + 4], cnt[q]);
    }
    __syncthreads();

    for (int i = threadIdx.x; i < M * 5; i += blockDim.x) {
        const int q = i / 5, f = i % 5;
        const float v = s_acc[i];
        if (f < 4) atomicAdd(&sums[(bi * M + q) * 4 + f], v);
        else       atomicAdd(&counts[bi * M + q], v);
    }
}

__global__ void il_means(const float* __restrict__ sums, const float* __restrict__ counts,
                         float* __restrict__ means) {
    const int i = threadIdx.x;
    if (i < B * M) {
        const float c = fmaxf(counts[i], 1.0f);
#pragma unroll
        for (int ch = 0; ch < C; ++ch) means[i * 4 + ch] = sums[i * 4 + ch] / c;
    }
}

#define HINGE_PIX(lv, v0, v1, v2, v3) {                                 \
        const float4 mu = s_mean[(lv)];                                 \
        const float d0 = (v0) - mu.x, d1 = (v1) - mu.y;                 \
        const float d2 = (v2) - mu.z, d3 = (v3) - mu.w;                 \
        const float dist = sqrtf(d0*d0 + d1*d1 + d2*d2 + d3*d3 + EPS_L);\
        float hi = fmaxf(dist - DELTA_VAR, 0.0f); hi *= hi;             \
        _Pragma("unroll")                                               \
        for (int q = 0; q < M; ++q) h[q] += ((lv) == q) ? hi : 0.0f;    \
    }

// Pass 3: per-pixel hinge distance to own mean (means staged into LDS by the
// Tensor Data Mover), reduced per (batch,label).
__global__ void il_pass3(const float* __restrict__ x, const int* __restrict__ lab,
                         const float* __restrict__ means, float* __restrict__ hsum) {
    const int bi  = blockIdx.x / BPB;
    const int blk = blockIdx.x % BPB;

    __shared__ float4 s_mean[M];
    __shared__ float  s_h[M];
    if (threadIdx.x < M) s_h[threadIdx.x] = 0.0f;

    // Stage this image's 9 x float4 means table into LDS with a 1-D TDM copy
    // (D#: count=1, type=2, data_size=4B, tensor_dim0=tile_dim0=36 dwords).
    // TDM ignores EXEC -> issue from wave 0 only, then barrier.
    if (threadIdx.x == 0) {
        const unsigned long long ga = (unsigned long long)(const void*)(means + (size_t)bi * M * 4);
        // Flat->LDS mapping truncates to addr[31:0]; use low bits as LDS byte addr.
        const unsigned int lds_addr = (unsigned int)(unsigned long long)(uintptr_t)&s_mean[0];
        u32x4 g0 = {0u, 0u, 0u, 0u};
        g0[0] = 1u;                                            // count=1 (valid D#)
        g0[1] = lds_addr;                                      // lds_addr [63:32]
        g0[2] = (unsigned int)(ga & 0xffffffffull);            // global_addr lo
        g0[3] = (unsigned int)((ga >> 32) & 0x01ffffffull)     // global_addr hi
              | (2u << 30);                                    // type=2 (image)
        i32x8 g1 = {0, 0, 0, 0, 0, 0, 0, 0};
        g1[0] = (2 << 16);                                     // data_size=4B
        g1[1] = (int)(36u << 16);                              // tensor_dim0=36
        g1[2] = (1 << 16);                                     // tensor_dim1=1
        g1[3] = (int)(36u << 16);                              // tile_dim0=36
        g1[4] = 1;                                             // tile_dim1=1
        g1[5] = 36;                                            // tensor_dim0_stride
        i32x4 gz = {0, 0, 0, 0};
#if defined(__clang_major__) && (__clang_major__ >= 23)
        i32x8 gz8 = {0, 0, 0, 0, 0, 0, 0, 0};
        __builtin_amdgcn_tensor_load_to_lds(g0, g1, gz, gz, gz8, 0);
#else
        __builtin_amdgcn_tensor_load_to_lds(g0, g1, gz, gz, 0);
#endif
        __builtin_amdgcn_s_wait_tensorcnt(0);
    }
    __syncthreads();

    float h[M];
#pragma unroll
    for (int q = 0; q < M; ++q) h[q] = 0.0f;

    const float4* xb4 = (const float4*)(x + (size_t)bi * C * NPIX);
    const int4*   lb4 = (const int4*)(lab + (size_t)bi * NPIX);
    const int stride = BPB * TPB;

    for (int p = blk * TPB + threadIdx.x; p < NPIX4; p += stride) {
        __builtin_prefetch(lb4 + p + stride, 0, 0);
        __builtin_prefetch(xb4 + p + stride, 0, 0);
        const int4   l4 = lb4[p];
        const float4 c0 = xb4[p];
        const float4 c1 = xb4[p + NPIX4];
        const float4 c2 = xb4[p + 2 * NPIX4];
        const float4 c3 = xb4[p + 3 * NPIX4];
        HINGE_PIX(l4.x, c0.x, c1.x, c2.x, c3.x)
        HINGE_PIX(l4.y, c0.y, c1.y, c2.y, c3.y)
        HINGE_PIX(l4.z, c0.z, c1.z, c2.z, c3.z)
        HINGE_PIX(l4.w, c0.w, c1.w, c2.w, c3.w)
    }

#pragma unroll
    for (int q = 0; q < M; ++q) atomicAdd(&s_h[q], h[q]);
    __syncthreads();
    if (threadIdx.x < M) atomicAdd(&hsum[bi * M + threadIdx.x], s_h[threadIdx.x]);
}

// Finalize: 8 waves (one per batch). Gram matrix of the 9x4 means via one
// V_WMMA_F32_16X16X4_F32 per wave; pairwise hinge + variance term + batch mean.
__global__ void il_finalize(const float* __restrict__ means, const float* __restrict__ counts,
                            const float* __restrict__ hsum, float* __restrict__ out) {
    __shared__ float s_norm[B][16];
    __shared__ float s_valid[B][16];
    __shared__ float s_var[B][16];
    __shared__ float s_lossd[B];

    const int wave = threadIdx.x >> 5;
    const int lane = threadIdx.x & 31;
    const int bi   = wave;

    if (lane < 16) {
        float nrm = 0.0f, vld = 0.0f, vr = 0.0f;
        if (lane < M) {
            const float  cntv = counts[bi * M + lane];
            const float4 mu   = ((const float4*)means)[bi * M + lane];
            nrm = mu.x * mu.x + mu.y * mu.y + mu.z * mu.z + mu.w * mu.w;
            const bool ok = (cntv > 1.0f) && (lane >= 1);
            vld = ok ? 1.0f : 0.0f;
            vr  = ok ? (hsum[bi * M + lane] / fmaxf(cntv, 1.0f)) : 0.0f;
        }
        s_norm[wave][lane]  = nrm;
        s_valid[wave][lane] = vld;
        s_var[wave][lane]   = vr;
    }
    if (lane == 0) s_lossd[wave] = 0.0f;
    __syncthreads();

    // A (16x4 f32): lane m = lane&15 holds row m; VGPR0 = K{0|2}, VGPR1 = K{1|3}
    // by lane half. B = A^T has the identical per-lane values (col n = lane&15),
    // so we feed the same registers for A and B -> D = means . means^T (Gram).
    const int m    = lane & 15;
    const int base = (lane >> 4) * 2;
    v2f a;
    a.x = (m < M) ? means[(bi * M + m) * 4 + base]     : 0.0f;
    a.y = (m < M) ? means[(bi * M + m) * 4 + base + 1] : 0.0f;
    v8f c = {};
    c = __builtin_amdgcn_wmma_f32_16x16x4_f32(false, a, false, a, (short)0, c, false, false);

    // D layout: lane half/col j = lane&15; VGPR r holds row i = r + 8*half.
    const int j    = lane & 15;
    const int half = lane >> 4;
    float acc = 0.0f;
#pragma unroll
    for (int r = 0; r < 8; ++r) {
        const int i = r + 8 * half;
        if (i < j) {
            const float wv = s_valid[wave][i] * s_valid[wave][j];
            const float g  = fmaxf(s_norm[wave][i] + s_norm[wave][j] - 2.0f * c[r], 0.0f);
            const float pd = sqrtf(g + EPS_L);
            const float t  = fmaxf(2.0f * DELTA_VAR - pd, 0.0f);
            acc += wv * t * t;
        }
    }
    atomicAdd(&s_lossd[wave], acc);
    __syncthreads();

    if (threadIdx.x == 0) {
        float tot = 0.0f;
        for (int bb = 0; bb < B; ++bb) {
            float K = 0.0f, lv = 0.0f;
            for (int l = 0; l < M; ++l) { K += s_valid[bb][l]; lv += s_var[bb][l]; }
            const float ld = (K > 1.0f) ? (s_lossd[bb] / (K - 1.0f + EPS_L)) : 0.0f;
            tot += lv + ld;
        }
        out[0] = tot / (float)B;
    }
}

extern "C" void kernel_launch(void* const* d_in, const int* in_sizes, int n_in,
                              void* d_out, int out_size, void* d_ws, size_t ws_size,
                              hipStream_t stream) {
    const float* x   = (const float*)d_in[0];
    const int*   lab = (const int*)d_in[1];
    float* out = (float*)d_out;
    float* ws  = (float*)d_ws;

    float* sums   = ws + WS_SUMS;
    float* counts = ws + WS_COUNTS;
    float* hsumv  = ws + WS_HSUM;
    float* means  = ws + WS_MEANS;

    il_zero_ws<<<1, 256, 0, stream>>>(ws);
    il_pass1<<<B * BPB, TPB, 0, stream>>>(x, lab, sums, counts);
    il_means<<<1, 128, 0, stream>>>(sums, counts, means);
    il_pass3<<<B * BPB, TPB, 0, stream>>>(x, lab, means, hsumv);
    il_finalize<<<1, B * 32, 0, stream>>>(means, counts, hsumv, out);
    (void)in_sizes; (void)n_in; (void)out_size; (void)ws_size;
}